// MultiHeadAttention_17446157157071
// MI455X (gfx1250) — compile-verified
//
#include <hip/hip_runtime.h>
#include <hip/hip_bf16.h>
#include <stdint.h>

// ---------------------------------------------------------------------------
// Autoformer AutoCorrelation on MI455X (gfx1250).
// FFT cross-correlation re-expressed as per-batch Gram GEMM + diagonal sums,
// so every heavy op is a WMMA f32 16x16x4 GEMM (fp32-exact).
// Global->LDS tile staging uses CDNA5 async copies (ASYNCcnt-tracked),
// double-buffered so the DMA of stage k+1 overlaps the WMMAs of stage k.
// ---------------------------------------------------------------------------

typedef __attribute__((ext_vector_type(2))) float v2f;
typedef __attribute__((ext_vector_type(8))) float v8f;

constexpr int Bsz  = 16;
constexpr int Lsz  = 1024;
constexpr int Dsz  = 512;
constexpr int Hh   = 8;
constexpr int DVv  = 64;
constexpr int HE   = Hh * DVv;   // 512
constexpr int TOPK = 6;

#define BDIM 256

__device__ __forceinline__ void async_copy16(unsigned lds_off, const float* gsrc) {
  // GLOBAL_LOAD_ASYNC_TO_LDS_B128: 16 contiguous bytes global -> LDS per lane,
  // no VGPR bounce, tracked by ASYNCcnt.
  asm volatile("global_load_async_to_lds_b128 %0, %1, off"
               :: "v"(lds_off), "v"(gsrc) : "memory");
}

__device__ __forceinline__ void wait_async_all() {
  asm volatile("s_wait_asynccnt 0" ::: "memory");
}

// ---------------------------------------------------------------------------
// Generic row-major GEMM: C[M,N] = A[M,K] * B[K,N]
// Block tile 64x64 (M x N), 8 waves; each wave owns a 16x32 strip =
// two independent 16x16 WMMA accumulators (breaks the WMMA RAW chain).
// K staged through double-buffered LDS in chunks of 32 via async b128 copies.
// ---------------------------------------------------------------------------
__global__ __launch_bounds__(BDIM) void gemm_f32(const float* __restrict__ A,
                                                 const float* __restrict__ Bm,
                                                 float* __restrict__ C,
                                                 int M, int N, int K) {
  __shared__ float As[2][64][36];   // row pitch 144 B (16B multiple)
  __shared__ float Bs[2][32][68];   // row pitch 272 B (16B multiple)
  const int tid  = threadIdx.x;
  const int lane = tid & 31;
  const int wave = tid >> 5;
  const int bm = blockIdx.x * 64;
  const int bn = blockIdx.y * 64;
  const int wm = (wave & 3) * 16;   // 0,16,32,48
  const int wn = (wave >> 2) * 32;  // 0,32

  auto stage = [&](int buf, int kt) {
    // A tile: 64 rows x 32 K (512 groups of 4 floats)
    for (int g = tid; g < 512; g += BDIM) {
      int r = g >> 3, c = (g & 7) << 2;
      async_copy16((unsigned)(uintptr_t)&As[buf][r][c],
                   &A[(size_t)(bm + r) * K + (kt + c)]);
    }
    // B tile: 32 K x 64 cols (512 groups of 4 floats, contiguous along N)
    for (int g = tid; g < 512; g += BDIM) {
      int r = g >> 4, c = (g & 15) << 2;
      async_copy16((unsigned)(uintptr_t)&Bs[buf][r][c],
                   &Bm[(size_t)(kt + r) * N + (bn + c)]);
    }
  };

  v8f acc0 = {}, acc1 = {};
  int cur = 0;
  stage(0, 0);
  for (int kt = 0; kt < K; kt += 32) {
    wait_async_all();
    __syncthreads();                       // buf[cur] ready; buf[cur^1] free
    if (kt + 32 < K) stage(cur ^ 1, kt + 32);

    const int rowA = wm + (lane & 15);
    const int c0   = wn + (lane & 15);
    const int c1   = c0 + 16;
    const int kh   = (lane >> 4) << 1;  // lanes 0-15 -> K0,K1 ; 16-31 -> K2,K3
#pragma unroll
    for (int k4 = 0; k4 < 32; k4 += 4) {
      v2f a  = *(const v2f*)&As[cur][rowA][k4 + kh];   // ds_load_b64
      v2f b0, b1;
      b0.x = Bs[cur][k4 + kh][c0];
      b0.y = Bs[cur][k4 + kh + 1][c0];
      b1.x = Bs[cur][k4 + kh][c1];
      b1.y = Bs[cur][k4 + kh + 1][c1];
      acc0 = __builtin_amdgcn_wmma_f32_16x16x4_f32(false, a, false, b0,
                                                   (short)0, acc0, false, false);
      acc1 = __builtin_amdgcn_wmma_f32_16x16x4_f32(false, a, false, b1,
                                                   (short)0, acc1, false, false);
    }
    cur ^= 1;
  }

  const int rbase = bm + wm + ((lane >> 4) << 3);
  const int col0  = bn + wn + (lane & 15);
#pragma unroll
  for (int i = 0; i < 8; ++i) {
    C[(size_t)(rbase + i) * N + col0]      = acc0[i];
    C[(size_t)(rbase + i) * N + col0 + 16] = acc1[i];
  }
}

// ---------------------------------------------------------------------------
// Per-batch Gram GEMM G = q_s[b] @ k_s[b]^T with fused diagonal reduction:
//   mean_value[b, (t_q - t_k) mod L] += G[t_q, t_k] / (H*E)
// B tile is stored K-contiguous per column (BsT[n][k]) so the async copy
// moves 16 contiguous bytes on both sides AND fragment loads are ds_load_b64.
// ---------------------------------------------------------------------------
__global__ __launch_bounds__(BDIM) void gram_corr(const float* __restrict__ qs,
                                                  const float* __restrict__ ks,
                                                  float* __restrict__ mean_value) {
  __shared__ float As[2][64][36];
  __shared__ float BsT[2][64][36];   // [t_k local][K local]
  const int tid  = threadIdx.x;
  const int lane = tid & 31;
  const int wave = tid >> 5;
  const int b  = blockIdx.z;
  const int bm = blockIdx.x * 64;   // t_q tile
  const int bn = blockIdx.y * 64;   // t_k tile
  const int wm = (wave & 3) * 16;
  const int wn = (wave >> 2) * 32;

  const float* Ap = qs + (size_t)b * Lsz * HE;
  const float* Bp = ks + (size_t)b * Lsz * HE;

  auto stage = [&](int buf, int kt) {
    for (int g = tid; g < 512; g += BDIM) {
      int r = g >> 3, c = (g & 7) << 2;
      async_copy16((unsigned)(uintptr_t)&As[buf][r][c],
                   &Ap[(size_t)(bm + r) * HE + (kt + c)]);
    }
    for (int g = tid; g < 512; g += BDIM) {
      int n = g >> 3, c = (g & 7) << 2;   // K is the fast (contiguous) axis
      async_copy16((unsigned)(uintptr_t)&BsT[buf][n][c],
                   &Bp[(size_t)(bn + n) * HE + (kt + c)]);
    }
  };

  v8f acc0 = {}, acc1 = {};
  int cur = 0;
  stage(0, 0);
  for (int kt = 0; kt < HE; kt += 32) {
    wait_async_all();
    __syncthreads();
    if (kt + 32 < HE) stage(cur ^ 1, kt + 32);

    const int rowA = wm + (lane & 15);
    const int c0   = wn + (lane & 15);
    const int c1   = c0 + 16;
    const int kh   = (lane >> 4) << 1;
#pragma unroll
    for (int k4 = 0; k4 < 32; k4 += 4) {
      v2f a  = *(const v2f*)&As[cur][rowA][k4 + kh];    // ds_load_b64
      v2f b0 = *(const v2f*)&BsT[cur][c0][k4 + kh];     // ds_load_b64
      v2f b1 = *(const v2f*)&BsT[cur][c1][k4 + kh];     // ds_load_b64
      acc0 = __builtin_amdgcn_wmma_f32_16x16x4_f32(false, a, false, b0,
                                                   (short)0, acc0, false, false);
      acc1 = __builtin_amdgcn_wmma_f32_16x16x4_f32(false, a, false, b1,
                                                   (short)0, acc1, false, false);
    }
    cur ^= 1;
  }

  const int tq0  = bm + wm + ((lane >> 4) << 3);
  const int tk0  = bn + wn + (lane & 15);
  const int tk1  = tk0 + 16;
  const float sc = 1.0f / (float)HE;
  float* mv = mean_value + (size_t)b * Lsz;
#pragma unroll
  for (int i = 0; i < 8; ++i) {
    atomicAdd(&mv[(tq0 + i - tk0) & (Lsz - 1)], acc0[i] * sc);
    atomicAdd(&mv[(tq0 + i - tk1) & (Lsz - 1)], acc1[i] * sc);
  }
}

__global__ void zero_f32(float* __restrict__ p, int n) {
  int i = blockIdx.x * blockDim.x + threadIdx.x;
  if (i < n) p[i] = 0.0f;
}

// ---------------------------------------------------------------------------
// Batch-mean over mean_value, iterative top-6 argmax (ties -> lower index,
// matching lax.top_k), then per-batch softmax of selected weights.
// ---------------------------------------------------------------------------
__global__ __launch_bounds__(BDIM) void topk_softmax(const float* __restrict__ mean_value,
                                                     int* __restrict__ idx_out,
                                                     float* __restrict__ w_out) {
  __shared__ float sm[Lsz];
  __shared__ float sv[BDIM];
  __shared__ int   si[BDIM];
  __shared__ int   sel[TOPK];
  const int tid = threadIdx.x;

  for (int l = tid; l < Lsz; l += BDIM) {
    float s = 0.0f;
    for (int b = 0; b < Bsz; ++b) s += mean_value[b * Lsz + l];
    sm[l] = s * (1.0f / (float)Bsz);
  }
  __syncthreads();

  for (int k = 0; k < TOPK; ++k) {
    float best = -__builtin_inff();
    int   bi   = 0x7fffffff;
    for (int l = tid; l < Lsz; l += BDIM) {
      float v = sm[l];
      if (v > best || (v == best && l < bi)) { best = v; bi = l; }
    }
    sv[tid] = best; si[tid] = bi;
    __syncthreads();
    for (int off = BDIM / 2; off > 0; off >>= 1) {
      if (tid < off) {
        float v = sv[tid + off]; int j = si[tid + off];
        if (v > sv[tid] || (v == sv[tid] && j < si[tid])) { sv[tid] = v; si[tid] = j; }
      }
      __syncthreads();
    }
    if (tid == 0) { sel[k] = si[0]; sm[si[0]] = -__builtin_inff(); }
    __syncthreads();
  }

  if (tid < TOPK) idx_out[tid] = sel[tid];

  if (tid < Bsz) {
    float w[TOPK];
    float mx = -__builtin_inff();
    for (int k = 0; k < TOPK; ++k) {
      w[k] = mean_value[tid * Lsz + sel[k]];
      mx = fmaxf(mx, w[k]);
    }
    float s = 0.0f;
    for (int k = 0; k < TOPK; ++k) { w[k] = __expf(w[k] - mx); s += w[k]; }
    float inv = 1.0f / s;
    for (int k = 0; k < TOPK; ++k) w_out[tid * TOPK + k] = w[k] * inv;
  }
}

// ---------------------------------------------------------------------------
// context[b,l,c] = sum_k w[b,k] * v_s[b, (l + delay_k) mod L, c]
// ---------------------------------------------------------------------------
__global__ __launch_bounds__(BDIM) void blend_rolls(const float* __restrict__ vs,
                                                    const int* __restrict__ idx,
                                                    const float* __restrict__ w,
                                                    float* __restrict__ ctx) {
  int gid = blockIdx.x * BDIM + threadIdx.x;
  const int total = Bsz * Lsz * HE;
  if (gid >= total) return;
  int c = gid % HE;
  int l = (gid / HE) % Lsz;
  int b = gid / (HE * Lsz);
  float s = 0.0f;
#pragma unroll
  for (int k = 0; k < TOPK; ++k) {
    int src = (l + idx[k]) & (Lsz - 1);
    s += w[b * TOPK + k] * vs[((size_t)b * Lsz + src) * HE + c];
  }
  ctx[gid] = s;
}

// ---------------------------------------------------------------------------
extern "C" void kernel_launch(void* const* d_in, const int* in_sizes, int n_in,
                              void* d_out, int out_size, void* d_ws, size_t ws_size,
                              hipStream_t stream) {
  (void)in_sizes; (void)n_in; (void)out_size; (void)ws_size;

  const float* Q   = (const float*)d_in[0];
  const float* K   = (const float*)d_in[1];
  const float* V   = (const float*)d_in[2];
  // d_in[3] = attn_mask (bool), ignored by AutoCorrelation
  const float* WQ  = (const float*)d_in[4];
  const float* WK  = (const float*)d_in[5];
  const float* WV  = (const float*)d_in[6];
  const float* Wfc = (const float*)d_in[7];
  float* out = (float*)d_out;

  char* ws = (char*)d_ws;
  const size_t matBytes = (size_t)Bsz * Lsz * HE * sizeof(float);  // 32 MB
  float* qs  = (float*)(ws);
  float* ks  = (float*)(ws + matBytes);
  float* vs  = (float*)(ws + 2 * matBytes);
  float* mv  = (float*)(ws + 3 * matBytes);                  // [B, L]
  int*   idx = (int*)  (ws + 3 * matBytes + (size_t)Bsz * Lsz * sizeof(float));
  float* tw  = (float*)((char*)idx + 64);                    // [B, TOPK]
  float* ctx = qs;  // q_s is dead after gram_corr; reuse for context

  const int M = Bsz * Lsz;  // 16384
  dim3 blk(BDIM);
  dim3 gProj(M / 64, Dsz / 64);          // 256 x 8

  // Input projections (Q/K/V share the same GEMM shape)
  gemm_f32<<<gProj, blk, 0, stream>>>(Q, WQ, qs, M, Dsz, Dsz);
  gemm_f32<<<gProj, blk, 0, stream>>>(K, WK, ks, M, Dsz, Dsz);
  gemm_f32<<<gProj, blk, 0, stream>>>(V, WV, vs, M, Dsz, Dsz);

  // mean_value accumulation target must start at zero
  zero_f32<<<(Bsz * Lsz + BDIM - 1) / BDIM, blk, 0, stream>>>(mv, Bsz * Lsz);

  // Per-batch Gram GEMM + diagonal reduction == FFT autocorrelation mean
  dim3 gGram(Lsz / 64, Lsz / 64, Bsz);   // 16 x 16 x 16
  gram_corr<<<gGram, blk, 0, stream>>>(qs, ks, mv);

  // Shared top-6 delays + per-batch softmax weights
  topk_softmax<<<1, blk, 0, stream>>>(mv, idx, tw);

  // Weighted blend of rolled V
  blend_rolls<<<(Bsz * Lsz * HE + BDIM - 1) / BDIM, blk, 0, stream>>>(vs, idx, tw, ctx);

  // Output projection
  gemm_f32<<<gProj, blk, 0, stream>>>(ctx, Wfc, out, M, Dsz, HE);
}